// MOPVRP_32461362823522
// MI455X (gfx1250) — compile-verified
//
#include <hip/hip_runtime.h>
#include <math.h>

#define Bn 64
#define Hn 128
#define Vn 10
#define Nn 1000
#define Wn 8
#define NHEADn 4
#define HDn 32

// d_out layout (flat, reference return order)
#define OUT_VP 0          // veh_probs  B*V      = 640
#define OUT_CP 640        // cust_probs B*N      = 64000
#define OUT_VI 64640      // veh_idx    B        = 64
#define OUT_HT 64704      // h_t        B*H      = 8192
#define OUT_CT 72896      // c_t        B*H      = 8192
#define OUT_HM 81088      // h_mem_new  B*W*H    = 65536

// workspace layout (floats)
#define WS_VREPR 0                      // B*V*H        = 81920
#define WS_HENR  81920                  // B*H          = 8192
#define WS_VOUT  90112                  // B*H          = 8192
#define WS_WVEC  98304                  // B*16*128     = 131072 (zero-padded A tiles)
#define WS_S0    229376                 // B*4          = 256
#define WS_VIDX  229632                 // B ints       = 64
#define WS_SC    229696                 // B*4*1024     = 262144
#define WS_LOG   491840                 // B*1024       = 65536   (end = 557376 floats)

#define INV_SQRT_HD 0.17677669529663687f

typedef __attribute__((ext_vector_type(2))) float v2f;
typedef __attribute__((ext_vector_type(8))) float v8f;

__device__ __forceinline__ float sigf(float x) { return 1.0f / (1.0f + expf(-x)); }

// ---------------- K1: veh_repr[b][v][h] = mean_n pairwise[b][h][v][n] ----------------
__global__ void k_mean(const float* __restrict__ pair, float* __restrict__ vrepr) {
    int wid  = blockIdx.x * (blockDim.x >> 5) + (threadIdx.x >> 5);
    int lane = threadIdx.x & 31;
    int b   = wid / (Vn * Hn);
    int rem = wid % (Vn * Hn);
    int v = rem / Hn;
    int h = rem % Hn;
    const float4* row = (const float4*)(pair + (((size_t)b * Hn + h) * Vn + v) * Nn);
    float s = 0.f;
    for (int i = lane; i < Nn / 4; i += 32) {  // 250 float4 per row, 16B aligned
        float4 x = row[i];
        s += x.x + x.y + x.z + x.w;
    }
    for (int off = 16; off > 0; off >>= 1) s += __shfl_xor(s, off, 32);
    if (lane == 0) vrepr[((size_t)b * Vn + v) * Hn + h] = s * (1.0f / (float)Nn);
}

// ---------------- K2: LSTM cell + h_mem shift ----------------
__global__ void k_lstm(const float* __restrict__ x, const float* __restrict__ h0,
                       const float* __restrict__ c0, const float* __restrict__ hmem,
                       const float* __restrict__ Wih, const float* __restrict__ Whh,
                       const float* __restrict__ bih, const float* __restrict__ bhh,
                       float* __restrict__ outbuf) {
    __shared__ float xs[Hn], hs[Hn], g[4 * Hn];
    int b = blockIdx.x, th = threadIdx.x;  // blockDim = 512
    if (th < Hn) { xs[th] = x[b * Hn + th]; hs[th] = h0[b * Hn + th]; }
    __syncthreads();
    float acc = bih[th] + bhh[th];
    const float* wi = Wih + (size_t)th * Hn;
    const float* wh = Whh + (size_t)th * Hn;
    for (int k = 0; k < Hn; k++) acc += xs[k] * wi[k] + hs[k] * wh[k];
    g[th] = acc;
    // shift h_mem: rows 0..6 of h_mem_new = h_mem rows 1..7
    float* hm = outbuf + OUT_HM + (size_t)b * Wn * Hn;
    for (int i = th; i < (Wn - 1) * Hn; i += 512)
        hm[i] = hmem[(size_t)b * Wn * Hn + Hn + i];
    __syncthreads();
    if (th < Hn) {
        float i_ = sigf(g[th]);
        float f_ = sigf(g[Hn + th]);
        float gg = tanhf(g[2 * Hn + th]);
        float o_ = sigf(g[3 * Hn + th]);
        float c2 = f_ * c0[b * Hn + th] + i_ * gg;
        float ht = o_ * tanhf(c2);
        outbuf[OUT_HT + b * Hn + th] = ht;
        outbuf[OUT_CT + b * Hn + th] = c2;
        hm[(Wn - 1) * Hn + th] = ht;
    }
}

// ---------------- K3: GRU over 8 steps + memory readout + h_enr ----------------
__global__ void k_gru(const float* __restrict__ Wih, const float* __restrict__ Whh,
                      const float* __restrict__ bih, const float* __restrict__ bhh,
                      const float* __restrict__ Wro, const float* __restrict__ bro,
                      const float* __restrict__ outbuf, float* __restrict__ henr) {
    __shared__ float hs[Hn], xs[Hn], gi[3 * Hn], gh[3 * Hn], hsum[Hn];
    int b = blockIdx.x, th = threadIdx.x;  // blockDim = 384
    if (th < Hn) { hs[th] = 0.f; hsum[th] = 0.f; }
    const float* hm = outbuf + OUT_HM + (size_t)b * Wn * Hn;
    for (int t = 0; t < Wn; t++) {
        __syncthreads();
        if (th < Hn) xs[th] = hm[t * Hn + th];
        __syncthreads();
        float ai = bih[th], ah = bhh[th];
        const float* wi = Wih + (size_t)th * Hn;
        const float* wh = Whh + (size_t)th * Hn;
        for (int k = 0; k < Hn; k++) { ai += xs[k] * wi[k]; ah += hs[k] * wh[k]; }
        gi[th] = ai; gh[th] = ah;
        __syncthreads();
        if (th < Hn) {
            float r = sigf(gi[th] + gh[th]);
            float z = sigf(gi[Hn + th] + gh[Hn + th]);
            float n = tanhf(gi[2 * Hn + th] + r * gh[2 * Hn + th]);
            float h2 = (1.0f - z) * n + z * hs[th];
            hs[th] = h2; hsum[th] += h2;
        }
    }
    __syncthreads();
    if (th < Hn) xs[th] = hsum[th] * (1.0f / (float)Wn);
    __syncthreads();
    if (th < Hn) {
        float a = bro[th];
        const float* w = Wro + (size_t)th * Hn;
        for (int k = 0; k < Hn; k++) a += xs[k] * w[k];
        henr[b * Hn + th] = outbuf[OUT_HT + b * Hn + th] + tanhf(a);
    }
}

// ---------------- K4: vehicle MHA + LN + veh_probs + argmax + cust-query fold ----------------
__global__ void k_veh(const float* __restrict__ vrepr, const float* __restrict__ henr,
                      const float* __restrict__ mveh,
                      const float* __restrict__ Wq, const float* __restrict__ Wk,
                      const float* __restrict__ Wv, const float* __restrict__ bq,
                      const float* __restrict__ bk, const float* __restrict__ bv,
                      const float* __restrict__ Wo, const float* __restrict__ bo,
                      const float* __restrict__ lng, const float* __restrict__ lnb,
                      const float* __restrict__ cWq, const float* __restrict__ cWk,
                      const float* __restrict__ cbq, const float* __restrict__ cbk,
                      float* __restrict__ outbuf, float* __restrict__ vout,
                      float* __restrict__ wvec, float* __restrict__ s0ws,
                      int* __restrict__ vidx) {
    __shared__ float he[Hn], q[Hn], kh[Vn * Hn], vh[Vn * Hn], oh[Hn], red[Hn];
    __shared__ float sc[NHEADn * Vn], att[NHEADn * Vn];
    __shared__ float mu_s, rstd_s;
    int b = blockIdx.x, th = threadIdx.x;  // blockDim = 128
    he[th] = henr[b * Hn + th];
    __syncthreads();
    // Q projection
    {
        float a = bq[th];
        const float* w = Wq + (size_t)th * Hn;
        for (int k = 0; k < Hn; k++) a += he[k] * w[k];
        q[th] = a;
    }
    // K/V projections for V=10 vehicles
    for (int v = 0; v < Vn; v++) {
        const float* vr = vrepr + ((size_t)b * Vn + v) * Hn;
        float ak = bk[th], av = bv[th];
        const float* wk = Wk + (size_t)th * Hn;
        const float* wv = Wv + (size_t)th * Hn;
        for (int k = 0; k < Hn; k++) { float x = vr[k]; ak += x * wk[k]; av += x * wv[k]; }
        kh[v * Hn + th] = ak; vh[v * Hn + th] = av;
    }
    __syncthreads();
    // scores (masked)
    if (th < NHEADn * Vn) {
        int v = th / NHEADn, hd = th % NHEADn;
        float s = 0.f;
        for (int d = 0; d < HDn; d++) s += q[hd * HDn + d] * kh[v * Hn + hd * HDn + d];
        float m = mveh[b * Vn + v];
        sc[hd * Vn + v] = (m == 0.0f) ? -1e9f : s * INV_SQRT_HD;
    }
    __syncthreads();
    // per-head softmax over V
    if (th < NHEADn) {
        float mx = -1e30f;
        for (int v = 0; v < Vn; v++) mx = fmaxf(mx, sc[th * Vn + v]);
        float se = 0.f, e[Vn];
        for (int v = 0; v < Vn; v++) { e[v] = expf(sc[th * Vn + v] - mx); se += e[v]; }
        for (int v = 0; v < Vn; v++) att[th * Vn + v] = e[v] / se;
    }
    __syncthreads();
    // attn @ V, concat heads
    {
        int hd = th >> 5;
        float s = 0.f;
        for (int v = 0; v < Vn; v++) s += att[hd * Vn + v] * vh[v * Hn + th];
        oh[th] = s;
    }
    __syncthreads();
    // output projection
    float o = bo[th];
    {
        const float* wo = Wo + (size_t)th * Hn;
        for (int k = 0; k < Hn; k++) o += oh[k] * wo[k];
    }
    // layer norm (two-pass)
    red[th] = o; __syncthreads();
    for (int s = 64; s > 0; s >>= 1) { if (th < s) red[th] += red[th + s]; __syncthreads(); }
    if (th == 0) mu_s = red[0] * (1.0f / (float)Hn);
    __syncthreads();
    float mu = mu_s;
    float d0 = o - mu;
    red[th] = d0 * d0; __syncthreads();
    for (int s = 64; s > 0; s >>= 1) { if (th < s) red[th] += red[th + s]; __syncthreads(); }
    if (th == 0) rstd_s = rsqrtf(red[0] * (1.0f / (float)Hn) + 1e-5f);
    __syncthreads();
    float vo = d0 * rstd_s * lng[th] + lnb[th];
    vout[b * Hn + th] = vo;
    // veh_probs + argmax (serial over V=10, thread 0)
    if (th == 0) {
        float lg[Vn];
        float mx = -1e30f;
        for (int v = 0; v < Vn; v++) {
            float s = 0.f;
            for (int hd = 0; hd < NHEADn; hd++) s += att[hd * Vn + v];
            float vw = s * 0.25f;
            float m = mveh[b * Vn + v];
            lg[v] = (m > 0.5f) ? logf(vw + 1e-8f) : -1e9f;
            mx = fmaxf(mx, lg[v]);
        }
        float se = 0.f;
        for (int v = 0; v < Vn; v++) { lg[v] = expf(lg[v] - mx); se += lg[v]; }
        int best = 0; float bp = -1.f;
        for (int v = 0; v < Vn; v++) {
            float p = lg[v] / se;
            outbuf[OUT_VP + b * Vn + v] = p;
            if (p > bp) { bp = p; best = v; }
        }
        outbuf[OUT_VI + b] = (float)best;
        vidx[b] = best;
    }
    // customer query:  q2 = LN(out) @ cWq.T + cbq,  then fold through cWk:
    // wvec[hd][hin] = sum_d q2[hd*32+d] * cWk[hd*32+d][hin];  s0[hd] = q2_hd . cbk_hd
    oh[th] = vo; __syncthreads();
    {
        float a2 = cbq[th];
        const float* w2 = cWq + (size_t)th * Hn;
        for (int k = 0; k < Hn; k++) a2 += oh[k] * w2[k];
        q[th] = a2;
    }
    __syncthreads();
    // zero-padded 16x128 A tile: rows 0..3 = folded head vectors, rows 4..15 = 0
    for (int hd = 0; hd < NHEADn; hd++) {
        float s = 0.f;
        for (int d = 0; d < HDn; d++) s += q[hd * HDn + d] * cWk[(size_t)(hd * HDn + d) * Hn + th];
        wvec[((size_t)b * 16 + hd) * Hn + th] = s;
    }
    for (int r = NHEADn; r < 16; r++) wvec[((size_t)b * 16 + r) * Hn + th] = 0.0f;
    if (th < NHEADn) {
        float s = 0.f;
        for (int d = 0; d < HDn; d++) s += q[th * HDn + d] * cbk[th * HDn + d];
        s0ws[b * NHEADn + th] = s;
    }
}

// ---------------- K5: WMMA customer scores: sel_b(1000x128) x wvec(128x16) ----------------
// One wave per (b, 16-row n-tile). A = zero-padded wvec tile (16x4 per step),
// B = 16 contiguous sel columns (4x16 per step).
// ISA f32 layouts: A vgpr0 holds K={0|2}, vgpr1 K={1|3} (lanes 0-15 | 16-31); B mirrors.
// No divergence in the loop: A is pre-padded, B uses a clamped index; validity is
// applied only at the store, so EXEC stays all-ones through every v_wmma.
__global__ void k_scores(const float* __restrict__ pair, const float* __restrict__ mpw,
                         const int* __restrict__ vidx, const float* __restrict__ wvec,
                         const float* __restrict__ s0ws, float* __restrict__ sc) {
    int wid  = blockIdx.x * (blockDim.x >> 5) + (threadIdx.x >> 5);
    int lane = threadIdx.x & 31;
    int b = wid / 63;           // 63 tiles of 16 cover N=1000 (padded)
    int t = wid % 63;
    int vi = vidx[b];
    const float* sel = pair + (size_t)b * Hn * Vn * Nn + (size_t)vi * Nn;  // sel(h,n)=sel[h*10000+n]
    const float* wv  = wvec + (size_t)b * 16 * Hn;
    int m    = lane & 15;               // A row
    int koff = (lane < 16) ? 0 : 2;     // K half per ISA 32-bit A/B layout
    int n    = lane & 15;               // B column within tile
    int gn   = t * 16 + n;
    bool valid = gn < Nn;
    int gnc  = valid ? gn : (Nn - 1);   // clamped load index: no exec divergence
    const float* arow = wv + m * Hn + koff;          // contiguous pair -> b64 load
    const float* bcol = sel + (size_t)koff * (Vn * Nn) + gnc;
    v8f c = {};
#pragma unroll 8
    for (int kk = 0; kk < 32; kk++) {   // K = 128 in steps of 4
        int h0 = kk * 4;
        v2f a  = *(const v2f*)(arow + h0);           // wv[m][h0+koff], wv[m][h0+koff+1]
        v2f bm;
        bm.x = bcol[(size_t)h0 * (Vn * Nn)];
        bm.y = bcol[(size_t)(h0 + 1) * (Vn * Nn)];
        c = __builtin_amdgcn_wmma_f32_16x16x4_f32(false, a, false, bm, (short)0, c,
                                                  false, false);
    }
    // D: vgpr r, lanes 0-15 -> row M=r, col N=lane  => heads 0..3 live in c[0..3], lanes<16
    if (lane < 16 && valid) {
        float mk = mpw[((size_t)b * Vn + vi) * Nn + gn];
#pragma unroll
        for (int r = 0; r < NHEADn; r++) {
            float s = (c[r] + s0ws[b * NHEADn + r]) * INV_SQRT_HD;
            sc[(((size_t)b * NHEADn + r) << 10) + gn] = (mk == 0.0f) ? -1e9f : s;
        }
    }
}

// ---------------- K6: per-head softmax over N, head-mean, log-masked-softmax ----------------
__global__ void k_soft(const float* __restrict__ sc, const float* __restrict__ mpw,
                       const int* __restrict__ vidx, float* __restrict__ logits,
                       float* __restrict__ outbuf) {
    __shared__ float red[256];
    __shared__ float mx4[NHEADn], se4[NHEADn], Ms, Ss;
    int b = blockIdx.x, th = threadIdx.x;  // blockDim = 256
    const float* sb = sc + ((size_t)b * NHEADn << 10);
    for (int hd = 0; hd < NHEADn; hd++) {
        float m = -1e30f;
        for (int n = th; n < Nn; n += 256) m = fmaxf(m, sb[(hd << 10) + n]);
        red[th] = m; __syncthreads();
        for (int s = 128; s > 0; s >>= 1) { if (th < s) red[th] = fmaxf(red[th], red[th + s]); __syncthreads(); }
        if (th == 0) mx4[hd] = red[0];
        __syncthreads();
        float mm = mx4[hd], su = 0.f;
        for (int n = th; n < Nn; n += 256) su += expf(sb[(hd << 10) + n] - mm);
        red[th] = su; __syncthreads();
        for (int s = 128; s > 0; s >>= 1) { if (th < s) red[th] += red[th + s]; __syncthreads(); }
        if (th == 0) se4[hd] = red[0];
        __syncthreads();
    }
    int vi = vidx[b];
    const float* mrow = mpw + ((size_t)b * Vn + vi) * Nn;
    float* lg = logits + (size_t)b * 1024;
    float lmax = -1e30f;
    for (int n = th; n < Nn; n += 256) {
        float cw = 0.f;
        for (int hd = 0; hd < NHEADn; hd++) cw += expf(sb[(hd << 10) + n] - mx4[hd]) / se4[hd];
        cw *= 0.25f;
        float l = (mrow[n] > 0.5f) ? logf(cw + 1e-8f) : -1e9f;
        lg[n] = l;
        lmax = fmaxf(lmax, l);
    }
    red[th] = lmax; __syncthreads();
    for (int s = 128; s > 0; s >>= 1) { if (th < s) red[th] = fmaxf(red[th], red[th + s]); __syncthreads(); }
    if (th == 0) Ms = red[0];
    __syncthreads();
    float su = 0.f;
    for (int n = th; n < Nn; n += 256) su += expf(lg[n] - Ms);
    red[th] = su; __syncthreads();
    for (int s = 128; s > 0; s >>= 1) { if (th < s) red[th] += red[th + s]; __syncthreads(); }
    if (th == 0) Ss = red[0];
    __syncthreads();
    float inv = 1.0f / Ss;
    for (int n = th; n < Nn; n += 256)
        outbuf[OUT_CP + b * Nn + n] = expf(lg[n] - Ms) * inv;
}

extern "C" void kernel_launch(void* const* d_in, const int* in_sizes, int n_in,
                              void* d_out, int out_size, void* d_ws, size_t ws_size,
                              hipStream_t stream) {
    const float* pair   = (const float*)d_in[0];   // B*H*V*N
    const float* dec    = (const float*)d_in[1];   // B*H
    const float* h0     = (const float*)d_in[2];   // B*H
    const float* c0     = (const float*)d_in[3];   // B*H
    const float* hmem   = (const float*)d_in[4];   // B*W*H
    const float* mveh   = (const float*)d_in[5];   // B*V
    const float* mpw    = (const float*)d_in[6];   // B*V*N
    const float* lWih   = (const float*)d_in[7];
    const float* lWhh   = (const float*)d_in[8];
    const float* lbih   = (const float*)d_in[9];
    const float* lbhh   = (const float*)d_in[10];
    const float* gWih   = (const float*)d_in[11];
    const float* gWhh   = (const float*)d_in[12];
    const float* gbih   = (const float*)d_in[13];
    const float* gbhh   = (const float*)d_in[14];
    const float* roW    = (const float*)d_in[15];
    const float* rob    = (const float*)d_in[16];
    const float* vWq    = (const float*)d_in[17];
    const float* vWk    = (const float*)d_in[18];
    const float* vWv    = (const float*)d_in[19];
    const float* vbq    = (const float*)d_in[20];
    const float* vbk    = (const float*)d_in[21];
    const float* vbv    = (const float*)d_in[22];
    const float* vWo    = (const float*)d_in[23];
    const float* vbo    = (const float*)d_in[24];
    const float* cWq    = (const float*)d_in[25];
    const float* cWk    = (const float*)d_in[26];
    /* cust_Wv d_in[27], cust_bv d_in[30], cust_Wo d_in[31], cust_bo d_in[32] unused:
       the reference discards the customer-MHA output, only attention weights survive. */
    const float* cbq    = (const float*)d_in[28];
    const float* cbk    = (const float*)d_in[29];
    const float* lng    = (const float*)d_in[33];
    const float* lnb    = (const float*)d_in[34];

    float* out = (float*)d_out;
    float* ws  = (float*)d_ws;

    float* vrepr  = ws + WS_VREPR;
    float* henr   = ws + WS_HENR;
    float* vout   = ws + WS_VOUT;
    float* wvec   = ws + WS_WVEC;
    float* s0ws   = ws + WS_S0;
    int*   vidx   = (int*)(ws + WS_VIDX);
    float* scbuf  = ws + WS_SC;
    float* logits = ws + WS_LOG;

    // K1: 81920 row-reductions, 8 waves/block
    k_mean<<<(Bn * Vn * Hn) / 8, 256, 0, stream>>>(pair, vrepr);
    // K2: LSTM cell
    k_lstm<<<Bn, 512, 0, stream>>>(dec, h0, c0, hmem, lWih, lWhh, lbih, lbhh, out);
    // K3: GRU + readout
    k_gru<<<Bn, 384, 0, stream>>>(gWih, gWhh, gbih, gbhh, roW, rob, out, henr);
    // K4: vehicle MHA / LN / probs / argmax / customer-query fold
    k_veh<<<Bn, 128, 0, stream>>>(vrepr, henr, mveh, vWq, vWk, vWv, vbq, vbk, vbv,
                                  vWo, vbo, lng, lnb, cWq, cWk, cbq, cbk,
                                  out, vout, wvec, s0ws, vidx);
    // K5: WMMA customer scores: 64 batches * 63 tiles = 4032 waves, 8 waves/block
    k_scores<<<(Bn * 63) / 8, 256, 0, stream>>>(pair, mpw, vidx, wvec, s0ws, scbuf);
    // K6: softmaxes -> cust_probs
    k_soft<<<Bn, 256, 0, stream>>>(scbuf, mpw, vidx, logits, out);
}